// MPNDQN_79877801771102
// MI455X (gfx1250) — compile-verified
//
#include <hip/hip_runtime.h>
#include <math.h>

#define OBS_DIM    256
#define HIDDEN_DIM 512
#define ACTION_DIM 18
#define BATCH      1024
#define ETA        0.01f
#define LAMBDA     0.95f

typedef __attribute__((ext_vector_type(2))) float v2f;
typedef __attribute__((ext_vector_type(8))) float v8f;

// ---------------------------------------------------------------------------
// Kernel 1: base = obs @ W^T  via V_WMMA_F32_16X16X4_F32 (full f32 precision).
// One wave (32 lanes) computes one 16x16 output tile, stepping K by 4.
// A (obs tile, 16x4 MxK): lane = M (lane&15), VGPR0/1 = K = 2*(lane>>4)+{0,1}
// B (W tile,  4x16 KxN): lane = N (lane&15), VGPR0/1 = K = 2*(lane>>4)+{0,1}
// C/D 16x16 f32: VGPR i -> M = i + 8*(lane>>4), N = lane&15
// grid = (BATCH/16, HIDDEN/16), block = 32
// ---------------------------------------------------------------------------
__global__ void base_gemm_wmma(const float* __restrict__ obs,
                               const float* __restrict__ W,
                               float* __restrict__ base) {
  const int tileM = blockIdx.x * 16;   // batch rows
  const int tileN = blockIdx.y * 16;   // hidden cols
  const int lane  = threadIdx.x;
  const int half  = lane >> 4;         // 0 or 1
  const int l15   = lane & 15;

  const float* arow = obs + (size_t)(tileM + l15) * OBS_DIM + 2 * half;
  const float* brow = W   + (size_t)(tileN + l15) * OBS_DIM + 2 * half;

  v8f acc = {};
  #pragma unroll 4
  for (int k = 0; k < OBS_DIM; k += 4) {
    v2f a, b;
    a.x = arow[k];
    a.y = arow[k + 1];
    b.x = brow[k];
    b.y = brow[k + 1];
    acc = __builtin_amdgcn_wmma_f32_16x16x4_f32(
        /*neg_a=*/false, a, /*neg_b=*/false, b,
        /*c_mod=*/(short)0, acc, /*reuse_a=*/false, /*reuse_b=*/false);
  }

  #pragma unroll
  for (int i = 0; i < 8; ++i) {
    base[(size_t)(tileM + i + 8 * half) * HIDDEN_DIM + (tileN + l15)] = acc[i];
  }
}

// ---------------------------------------------------------------------------
// Kernel 2: fused plastic contraction + tanh + Hebbian update.
// Reads each M row exactly once, writes new_M exactly once (the entire
// bandwidth budget of the problem: ~1.07 GB at 23.3 TB/s).
// grid = (BATCH, 8), block = 256 (8 waves). Wave w of block (b, yb) handles
// h = yb*64 + it*8 + w for it in 0..7. Lanes split the 256-wide O dimension:
// lane owns obs/W/M elements [4*lane .. 4*lane+3] and [128+4*lane ..].
// ---------------------------------------------------------------------------
__global__ void mpn_fused(const float* __restrict__ obs,
                          const float* __restrict__ M,
                          const float* __restrict__ W,
                          const float* __restrict__ b_h,
                          const float* __restrict__ base,
                          float* __restrict__ hidden,
                          float* __restrict__ newM) {
  __shared__ float s_obs[OBS_DIM];
  const int b   = blockIdx.x;
  const int tid = threadIdx.x;
  if (tid < OBS_DIM) s_obs[tid] = obs[(size_t)b * OBS_DIM + tid];
  __syncthreads();

  const int wave = tid >> 5;
  const int lane = tid & 31;
  const int c0 = lane * 4;
  const int c1 = 128 + lane * 4;

  const float4 x0 = *(const float4*)(s_obs + c0);
  const float4 x1 = *(const float4*)(s_obs + c1);

  #pragma unroll 2
  for (int it = 0; it < 8; ++it) {
    const int h = blockIdx.y * 64 + it * 8 + wave;
    const size_t rowoff = ((size_t)b * HIDDEN_DIM + h) * OBS_DIM;

    // prefetch next iteration's M row for this wave (global_prefetch_b8)
    if (it + 1 < 8) {
      __builtin_prefetch(M + rowoff + (size_t)8 * OBS_DIM + c0, 0, 1);
    }

    const float4 m0 = *(const float4*)(M + rowoff + c0);
    const float4 m1 = *(const float4*)(M + rowoff + c1);
    const float4 w0 = *(const float4*)(W + (size_t)h * OBS_DIM + c0);
    const float4 w1 = *(const float4*)(W + (size_t)h * OBS_DIM + c1);

    // plastic partial: sum_o W*M*x over this lane's 8 elements
    float p = 0.0f;
    p = fmaf(w0.x * x0.x, m0.x, p);
    p = fmaf(w0.y * x0.y, m0.y, p);
    p = fmaf(w0.z * x0.z, m0.z, p);
    p = fmaf(w0.w * x0.w, m0.w, p);
    p = fmaf(w1.x * x1.x, m1.x, p);
    p = fmaf(w1.y * x1.y, m1.y, p);
    p = fmaf(w1.z * x1.z, m1.z, p);
    p = fmaf(w1.w * x1.w, m1.w, p);

    // wave32 butterfly reduction -> every lane holds the full sum
    #pragma unroll
    for (int off = 16; off > 0; off >>= 1) {
      p += __shfl_xor(p, off, 32);
    }

    const float hv =
        tanhf(base[(size_t)b * HIDDEN_DIM + h] + p + b_h[h]);
    if (lane == 0) hidden[(size_t)b * HIDDEN_DIM + h] = hv;

    const float s = ETA * hv;
    float4 n0, n1;
    n0.x = fmaf(LAMBDA, m0.x, s * x0.x);
    n0.y = fmaf(LAMBDA, m0.y, s * x0.y);
    n0.z = fmaf(LAMBDA, m0.z, s * x0.z);
    n0.w = fmaf(LAMBDA, m0.w, s * x0.w);
    n1.x = fmaf(LAMBDA, m1.x, s * x1.x);
    n1.y = fmaf(LAMBDA, m1.y, s * x1.y);
    n1.z = fmaf(LAMBDA, m1.z, s * x1.z);
    n1.w = fmaf(LAMBDA, m1.w, s * x1.w);
    *(float4*)(newM + rowoff + c0) = n0;
    *(float4*)(newM + rowoff + c1) = n1;
  }
}

// ---------------------------------------------------------------------------
// Kernel 3: q = hidden @ Wq^T + bq   (1024x18x512, trivial; Wq is L2-resident)
// grid = BATCH, block = 32, thread a computes q[b, a]
// ---------------------------------------------------------------------------
__global__ void q_head(const float* __restrict__ hidden,
                       const float* __restrict__ Wq,
                       const float* __restrict__ bq,
                       float* __restrict__ q) {
  const int b = blockIdx.x;
  const int a = threadIdx.x;
  if (a >= ACTION_DIM) return;
  const float* hrow = hidden + (size_t)b * HIDDEN_DIM;
  const float* wrow = Wq + (size_t)a * HIDDEN_DIM;
  float s = bq[a];
  #pragma unroll 4
  for (int h = 0; h < HIDDEN_DIM; h += 4) {
    const float4 hv = *(const float4*)(hrow + h);
    const float4 wv = *(const float4*)(wrow + h);
    s = fmaf(hv.x, wv.x, s);
    s = fmaf(hv.y, wv.y, s);
    s = fmaf(hv.z, wv.z, s);
    s = fmaf(hv.w, wv.w, s);
  }
  q[(size_t)b * ACTION_DIM + a] = s;
}

// ---------------------------------------------------------------------------
extern "C" void kernel_launch(void* const* d_in, const int* in_sizes, int n_in,
                              void* d_out, int out_size, void* d_ws, size_t ws_size,
                              hipStream_t stream) {
  const float* obs = (const float*)d_in[0];
  const float* M   = (const float*)d_in[1];
  const float* W   = (const float*)d_in[2];
  const float* b_h = (const float*)d_in[3];
  const float* Wq  = (const float*)d_in[4];
  const float* bq  = (const float*)d_in[5];

  float* q_out = (float*)d_out;                         // [1024, 18]
  float* newM  = q_out + (size_t)BATCH * ACTION_DIM;    // [1024, 512, 256]

  float* base   = (float*)d_ws;                         // [1024, 512]
  float* hidden = base + (size_t)BATCH * HIDDEN_DIM;    // [1024, 512]

  base_gemm_wmma<<<dim3(BATCH / 16, HIDDEN_DIM / 16), 32, 0, stream>>>(obs, W, base);
  mpn_fused<<<dim3(BATCH, 8), 256, 0, stream>>>(obs, M, W, b_h, base, hidden, newM);
  q_head<<<BATCH, 32, 0, stream>>>(hidden, Wq, bq, q_out);
}